// ModelNew_80908593922848
// MI455X (gfx1250) — compile-verified
//
#include <hip/hip_runtime.h>

typedef __attribute__((ext_vector_type(2))) float v2f;
typedef __attribute__((ext_vector_type(4))) float v4f;
typedef __attribute__((ext_vector_type(8))) float v8f;

#define CIN   32
#define COUT  64

#define LDSA_ROW   36                 // floats per iw row (16B-aligned, bank-spread)
#define LDSA_PLANE (32 * LDSA_ROW)    // 1152 floats per (id,ih) plane
#define LDSA_ZERO  (9 * LDSA_PLANE)   // offset of the 36-float zero row

// Branch-chosen int-punned float atomic max (correct for all sign combos with -inf init).
__device__ __forceinline__ void atomicMaxF(float* addr, float v) {
  if (v >= 0.0f) atomicMax((int*)addr, __float_as_int(v));
  else           atomicMin((unsigned int*)addr, __float_as_uint(v));
}

__global__ void init_neg_inf(float* p, int n) {
  int i = blockIdx.x * blockDim.x + threadIdx.x;
  if (i < n) p[i] = -__builtin_inff();
}

// weight [ci][co][kd][kh][kw] -> wTL[tap][hi][co][16], where the 16 floats are a
// lane's B operands for kb=0..7: element (kb,b) holds w[ci=4kb+2hi+b][co][tap].
__global__ void transpose_weight(const float* __restrict__ w, float* __restrict__ wTL) {
  int i = blockIdx.x * blockDim.x + threadIdx.x;
  if (i >= 125 * 2 * COUT * 16) return;
  int b   = i & 1;
  int kb  = (i >> 1) & 7;
  int co  = (i >> 4) & 63;
  int hi  = (i >> 10) & 1;
  int tap = i >> 11;
  int ci  = 4 * kb + 2 * hi + b;
  wTL[i] = w[ci * (COUT * 125) + co * 125 + tap];
}

// One workgroup per (n, od, oh) conv-output row: conv[co=0..63, ow=0..63] via f32 WMMA
// (M = ow parity-split, N = co, K = ci per tap), then 6-wide w-block max + atomic max.
__global__ __launch_bounds__(256)
void convt_pool_stage1(const float* __restrict__ x,
                       const float* __restrict__ wTL,
                       float* __restrict__ bm) {
  const int oh = blockIdx.x;   // 0..59
  const int od = blockIdx.y;   // 0..29
  const int n  = blockIdx.z;   // 0..15

  __shared__ __align__(16) float ldsA[LDSA_ZERO + LDSA_ROW];  // x planes + zero row
  __shared__ float conv_s[64 * 65];                           // [ow][co], pad-65

  const int odp = od & 1, ohp = oh & 1;
  const int ndt = odp ? 2 : 3;
  const int nht = ohp ? 2 : 3;

  if (threadIdx.x < LDSA_ROW) ldsA[LDSA_ZERO + threadIdx.x] = 0.0f;

  // ---- stage x planes into LDS, channel-permuted so A operands are contiguous ----
  const float* xn = x + (size_t)n * (CIN * 16 * 32 * 32);
  for (int di = 0; di < ndt; ++di) {
    int id = (od + 2 - (odp + 2 * di)) >> 1;
    if (id < 0 || id > 15) continue;
    for (int hj = 0; hj < nht; ++hj) {
      int ih = (oh + 2 - (ohp + 2 * hj)) >> 1;
      if (ih < 0 || ih > 31) continue;
      const float* src = xn + id * 1024 + ih * 32;
      float* dst = &ldsA[(di * 3 + hj) * LDSA_PLANE];
      for (int i = threadIdx.x; i < CIN * 32; i += 256) {
        int ci = i >> 5, iw = i & 31;
        int perm = ((ci >> 1) & 1) * 16 + (ci >> 2) * 2 + (ci & 1);
        dst[iw * LDSA_ROW + perm] = src[ci * 16384 + iw];
      }
    }
  }
  __syncthreads();

  // ---- wave-uniform control values forced into SGPRs ----
  const int wvu  = __builtin_amdgcn_readfirstlane(threadIdx.x >> 5);
  const int mtu  = wvu & 3;              // M-tile id (uniform per wave)
  const int n0   = (wvu >> 2) * 32;      // this wave's co range [n0, n0+32)
  const int paru = mtu >> 1;             // 0: even ow, 1: odd ow

  // per-lane values
  const int lane = threadIdx.x & 31;
  const int l16  = lane & 15;
  const int hi   = lane >> 4;
  const int mm   = (mtu & 1) * 16 + l16; // row in parity half (A-layout: M = lane%16)
  const int laneB = (hi * 64 + n0 + l16) * 16;   // lane-constant B offset (floats)

  v8f acc0 = {}, acc1 = {};

  // One tap: 2+2 ds_load_b128 (A) + 8 global_load_b128 (B) + 16 WMMA.
  auto tap_body = [&](int planeBase, int tapBase, int t) {
    const int iw = mm + 1 - t;                    // == (ow+2-kw)/2, parity-free
    const int rowOff = ((unsigned)iw < 32u) ? (planeBase + iw * LDSA_ROW)
                                            : LDSA_ZERO;          // zero row if OOB
    v2f A8[8];
    {
      const v4f* Ap = (const v4f*)&ldsA[rowOff + hi * 16];
      v4f* q = (v4f*)A8;
      q[0] = Ap[0]; q[1] = Ap[1]; q[2] = Ap[2]; q[3] = Ap[3];
    }
    v2f B0[8], B1[8];
    {
      const v4f* Bp0 = (const v4f*)(wTL + (tapBase + 2 * t) * 2048 + laneB);
      const v4f* Bp1 = Bp0 + 64;                  // co + 16 (256 floats ahead)
      v4f* q0 = (v4f*)B0;
      v4f* q1 = (v4f*)B1;
      q0[0] = Bp0[0]; q0[1] = Bp0[1]; q0[2] = Bp0[2]; q0[3] = Bp0[3];
      q1[0] = Bp1[0]; q1[1] = Bp1[1]; q1[2] = Bp1[2]; q1[3] = Bp1[3];
    }
    #pragma unroll
    for (int kb = 0; kb < 8; ++kb) {
      acc0 = __builtin_amdgcn_wmma_f32_16x16x4_f32(false, A8[kb], false, B0[kb],
                                                   (short)0, acc0, false, false);
      acc1 = __builtin_amdgcn_wmma_f32_16x16x4_f32(false, A8[kb], false, B1[kb],
                                                   (short)0, acc1, false, false);
    }
  };

  for (int di = 0; di < ndt; ++di) {
    int kd = odp + 2 * di;
    int id = (od + 2 - kd) >> 1;
    if (id < 0 || id > 15) continue;
    for (int hj = 0; hj < nht; ++hj) {
      int kh = ohp + 2 * hj;
      int ih = (oh + 2 - kh) >> 1;
      if (ih < 0 || ih > 31) continue;
      const int planeBase = (di * 3 + hj) * LDSA_PLANE;
      const int tapBase   = kd * 25 + kh * 5 + paru;   // kw = paru + 2t
      if (paru == 0) {                 // uniform scalar branch, constant-t bodies
        tap_body(planeBase, tapBase, 0);
        tap_body(planeBase, tapBase, 1);
        tap_body(planeBase, tapBase, 2);
      } else {
        tap_body(planeBase, tapBase, 0);
        tap_body(planeBase, tapBase, 1);
      }
    }
  }

  // ---- D-matrix layout: c[r] = D[M = r + 8*hi][N = l16] -> conv_s[ow][co] ----
  #pragma unroll
  for (int r = 0; r < 8; ++r) {
    int row = (mtu & 1) * 16 + r + 8 * hi;
    int oww = 2 * row + paru;
    conv_s[oww * 65 + n0 + l16]      = acc0[r];
    conv_s[oww * 65 + n0 + 16 + l16] = acc1[r];
  }
  __syncthreads();

  // ---- max over 6-wide w blocks (ow 0..59; 60..63 are pool-dropped) ----
  const int d6 = od / 6, h6 = oh / 6;
  for (int i = threadIdx.x; i < COUT * 10; i += 256) {
    int co = i / 10, wb = i % 10;
    float m = -__builtin_inff();
    #pragma unroll
    for (int j = 0; j < 6; ++j) m = fmaxf(m, conv_s[(wb * 6 + j) * 65 + co]);
    atomicMaxF(&bm[((size_t)(n * 64 + co) * 5 + d6) * 100 + h6 * 10 + wb], m);
  }
}

// out[n,0,d,h,w] = sum_co (blockmax + bias[co]); bias commutes with the max.
__global__ void sum_channels(const float* __restrict__ bm,
                             const float* __restrict__ bias,
                             float* __restrict__ out) {
  int idx = blockIdx.x * blockDim.x + threadIdx.x;
  if (idx >= 16 * 500) return;
  int n = idx / 500, s = idx % 500;
  float acc = 0.0f;
  for (int co = 0; co < 64; ++co)
    acc += bm[((size_t)(n * 64 + co)) * 500 + s] + bias[co];
  out[idx] = acc;
}

extern "C" void kernel_launch(void* const* d_in, const int* in_sizes, int n_in,
                              void* d_out, int out_size, void* d_ws, size_t ws_size,
                              hipStream_t stream) {
  const float* x    = (const float*)d_in[0];
  const float* w    = (const float*)d_in[1];
  const float* bias = (const float*)d_in[2];
  float* out = (float*)d_out;

  float* ws  = (float*)d_ws;
  float* bm  = ws;              // 16*64*5*10*10 = 512000 floats (2.0 MB)
  float* wTL = ws + 512000;     // 125*2*64*16  = 256000 floats (1.0 MB)

  init_neg_inf<<<2000, 256, 0, stream>>>(bm, 512000);
  transpose_weight<<<1000, 256, 0, stream>>>(w, wTL);
  convt_pool_stage1<<<dim3(60, 30, 16), 256, 0, stream>>>(x, wTL, bm);
  sum_channels<<<32, 256, 0, stream>>>(bm, bias, out);
}